// PrototypicalAttention_884763263839
// MI455X (gfx1250) — compile-verified
//
#include <hip/hip_runtime.h>
#include <math.h>

// Problem constants (from reference)
#define BB 16
#define NN 8
#define TT 8
#define CC 2048
#define HH 7
#define WW 7
#define PP 32
#define HWc 49            // H*W
#define THW 392           // T*H*W
#define ROWS (BB*NN)      // 128
#define EPSV 1e-12f
#define TEMP 0.07f

typedef __attribute__((ext_vector_type(2))) float v2f;
typedef __attribute__((ext_vector_type(8))) float v8f;

// ---------------------------------------------------------------------------
// K1: ff[bn][c] = mean over (t, h, w) of f[(bn*T+t), c, h, w]
// One wave per (bn, c). Lanes cover hw=0..48 (lane and lane+32), loop t=0..7.
// Fully coalesced 411MB stream -> bandwidth bound (~17.6us at 23.3 TB/s).
// ---------------------------------------------------------------------------
__global__ void k_reduce_f(const float* __restrict__ f, float* __restrict__ ff) {
    const int gtid = blockIdx.x * blockDim.x + threadIdx.x;
    const int wid  = gtid >> 5;              // global wave id = bn*2048 + c
    const int lane = threadIdx.x & 31;
    const int bn   = wid >> 11;              // / CC
    const int c    = wid & (CC - 1);

    const float* base = f + (size_t)bn * TT * CC * HWc + (size_t)c * HWc;
    float s = 0.0f;
#pragma unroll
    for (int t = 0; t < TT; ++t) {
        const float* p = base + (size_t)t * CC * HWc;
        s += p[lane];
        if (lane < HWc - 32) s += p[lane + 32];
    }
#pragma unroll
    for (int off = 16; off > 0; off >>= 1) s += __shfl_down(s, off);
    if (lane == 0) ff[wid] = s * (1.0f / (float)THW);
}

// ---------------------------------------------------------------------------
// K2: prototype L2 normalization. One 256-thread block per prototype row.
// ---------------------------------------------------------------------------
__global__ void k_pnorm(const float* __restrict__ proto, float* __restrict__ pn) {
    const int p   = blockIdx.x;
    const int tid = threadIdx.x;
    __shared__ float red[8];
    __shared__ float sInv;

    float s = 0.0f;
    for (int c = tid; c < CC; c += 256) {
        float v = proto[p * CC + c];
        s += v * v;
    }
#pragma unroll
    for (int off = 16; off > 0; off >>= 1) s += __shfl_down(s, off);
    if ((tid & 31) == 0) red[tid >> 5] = s;
    __syncthreads();
    if (tid == 0) {
        float t = 0.0f;
#pragma unroll
        for (int i = 0; i < 8; ++i) t += red[i];
        sInv = 1.0f / fmaxf(sqrtf(t), EPSV);
    }
    __syncthreads();
    const float inv = sInv;
    for (int c = tid; c < CC; c += 256) pn[p * CC + c] = proto[p * CC + c] * inv;
}

// ---------------------------------------------------------------------------
// K3: ff row norms (clamped to EPS). One block per row.
// ---------------------------------------------------------------------------
__global__ void k_ffnorm(const float* __restrict__ ff, float* __restrict__ rn) {
    const int r   = blockIdx.x;
    const int tid = threadIdx.x;
    __shared__ float red[8];
    float s = 0.0f;
    for (int c = tid; c < CC; c += 256) {
        float v = ff[r * CC + c];
        s += v * v;
    }
#pragma unroll
    for (int off = 16; off > 0; off >>= 1) s += __shfl_down(s, off);
    if ((tid & 31) == 0) red[tid >> 5] = s;
    __syncthreads();
    if (tid == 0) {
        float t = 0.0f;
#pragma unroll
        for (int i = 0; i < 8; ++i) t += red[i];
        rn[r] = fmaxf(sqrtf(t), EPSV);
    }
}

// ---------------------------------------------------------------------------
// K4: orthogonality loss. triu-sum of symmetric Gram = (||sum p_hat||^2 -
// sum ||p_hat_i||^2) / 2, normalized by P*(P-1)/2 = 496. One block.
// ---------------------------------------------------------------------------
__global__ void k_orth(const float* __restrict__ pn, float* __restrict__ out_loss) {
    const int tid = threadIdx.x;
    __shared__ float red1[8];
    __shared__ float red2[8];
    float a1 = 0.0f, a2 = 0.0f;
    for (int c = tid; c < CC; c += 256) {
        float q = 0.0f;
#pragma unroll
        for (int p = 0; p < PP; ++p) {
            float v = pn[p * CC + c];
            q  += v;
            a2 += v * v;
        }
        a1 += q * q;
    }
#pragma unroll
    for (int off = 16; off > 0; off >>= 1) {
        a1 += __shfl_down(a1, off);
        a2 += __shfl_down(a2, off);
    }
    if ((tid & 31) == 0) { red1[tid >> 5] = a1; red2[tid >> 5] = a2; }
    __syncthreads();
    if (tid == 0) {
        float t1 = 0.0f, t2 = 0.0f;
#pragma unroll
        for (int i = 0; i < 8; ++i) { t1 += red1[i]; t2 += red2[i]; }
        float triu = 0.5f * (t1 - t2);
        out_loss[0] = triu / (float)(PP * (PP - 1) / 2);   // LOSS_WEIGHT = 1
    }
}

// ---------------------------------------------------------------------------
// K5: similarity = (ff @ pn^T) / rownorm via V_WMMA_F32_16X16X4_F32.
// One 512-thread block = 16 waves; wave (mt, nt) owns one 16x16 output tile,
// K-loop of 512 wmma steps (K=4 each).
// A layout (f32 16x4): lane l holds M=l%16; v0 = K=(l<16?0:2), v1 = K+1.
// B layout (f32 4x16): lane l holds N=l%16; v0 = K=(l<16?0:2), v1 = K+1.
// C/D layout: VGPR r -> M = r + (l<16?0:8), N = l%16.
// ---------------------------------------------------------------------------
__global__ void k_sim_wmma(const float* __restrict__ ff, const float* __restrict__ pn,
                           const float* __restrict__ rn, float* __restrict__ sim) {
    const int wid  = threadIdx.x >> 5;   // 0..15
    const int lane = threadIdx.x & 31;
    const int mt   = wid >> 1;           // 0..7
    const int nt   = wid & 1;            // 0..1
    const int l16  = lane & 15;
    const int koff = (lane < 16) ? 0 : 2;

    const float* arow = ff + (size_t)(mt * 16 + l16) * CC + koff;  // 8B aligned
    const float* brow = pn + (size_t)(nt * 16 + l16) * CC + koff;  // 8B aligned

    v8f acc = {};
    for (int k = 0; k < CC; k += 4) {
        v2f a = *(const v2f*)(arow + k);
        v2f b = *(const v2f*)(brow + k);
        acc = __builtin_amdgcn_wmma_f32_16x16x4_f32(
            /*neg_a=*/false, a, /*neg_b=*/false, b,
            /*c_mod=*/(short)0, acc, /*reuse_a=*/false, /*reuse_b=*/false);
    }

    const int half = (lane < 16) ? 0 : 8;
#pragma unroll
    for (int r = 0; r < 8; ++r) {
        const int m = mt * 16 + r + half;
        sim[m * PP + nt * 16 + l16] = acc[r] / rn[m];
    }
}

// ---------------------------------------------------------------------------
// K6: per-batch max/argmax over P, softmax over N at T=0.07, affinity.
// One block per batch (16 blocks x 256 threads).
// ---------------------------------------------------------------------------
__global__ void k_final(const float* __restrict__ ff, const float* __restrict__ sim,
                        float* __restrict__ aff, int* __restrict__ amax) {
    const int b   = blockIdx.x;
    const int tid = threadIdx.x;
    __shared__ float ssim[NN * PP];   // 256 values
    __shared__ float attn[NN];
    __shared__ int   sarg[NN];

    ssim[tid] = sim[b * NN * PP + tid];
    __syncthreads();

    if (tid < NN) {
        float mx = ssim[tid * PP];
        int   am = 0;
#pragma unroll
        for (int p = 1; p < PP; ++p) {
            float v = ssim[tid * PP + p];
            if (v > mx) { mx = v; am = p; }   // first-max wins (matches argmax)
        }
        attn[tid] = mx;
        sarg[tid] = am;
    }
    __syncthreads();

    if (tid == 0) {
        float mx = attn[0];
#pragma unroll
        for (int n = 1; n < NN; ++n) mx = fmaxf(mx, attn[n]);
        float e[NN], s = 0.0f;
#pragma unroll
        for (int n = 0; n < NN; ++n) { e[n] = __expf((attn[n] - mx) * (1.0f / TEMP)); s += e[n]; }
        float inv = 1.0f / s;
#pragma unroll
        for (int n = 0; n < NN; ++n) attn[n] = e[n] * inv;
#pragma unroll
        for (int n = 0; n < NN; ++n) amax[b * NN + n] = sarg[n];
    }
    __syncthreads();

    for (int c = tid; c < CC; c += 256) {
        float s = 0.0f;
#pragma unroll
        for (int n = 0; n < NN; ++n) s += attn[n] * ff[(size_t)(b * NN + n) * CC + c];
        aff[b * CC + c] = s;
    }
}

// ---------------------------------------------------------------------------
// K7: histogram of 128 argmax indices -> frequency entropy. One block.
// ---------------------------------------------------------------------------
__global__ void k_entropy(const int* __restrict__ amax, float* __restrict__ outH) {
    __shared__ int cnt[PP];
    const int tid = threadIdx.x;  // 128 threads
    if (tid < PP) cnt[tid] = 0;
    __syncthreads();
    atomicAdd(&cnt[amax[tid]], 1);
    __syncthreads();
    if (tid == 0) {
        float H = 0.0f;
#pragma unroll
        for (int p = 0; p < PP; ++p) {
            float pf = (float)cnt[p] * (1.0f / (float)ROWS);
            H -= pf * logf(pf + EPSV);
        }
        outH[0] = H;
    }
}

// ---------------------------------------------------------------------------
extern "C" void kernel_launch(void* const* d_in, const int* in_sizes, int n_in,
                              void* d_out, int out_size, void* d_ws, size_t ws_size,
                              hipStream_t stream) {
    const float* f     = (const float*)d_in[0];
    const float* proto = (const float*)d_in[1];
    // d_in[2] (labels) unused by the reference outputs.

    float* out = (float*)d_out;              // [0..32767] affinity, [262144] H, [262145] loss
    float* ws  = (float*)d_ws;
    float* ff  = ws;                          // 128*2048
    float* pn  = ff + ROWS * CC;              // 32*2048
    float* rn  = pn + PP * CC;                // 128
    float* sim = rn + ROWS;                   // 128*32
    int*   am  = (int*)(sim + ROWS * PP);     // 128 ints

    // K1: 128*2048 waves, 8 waves per 256-thread block.
    k_reduce_f<<<(ROWS * CC) / 8, 256, 0, stream>>>(f, ff);
    k_pnorm  <<<PP,   256, 0, stream>>>(proto, pn);
    k_ffnorm <<<ROWS, 256, 0, stream>>>(ff, rn);
    k_orth   <<<1,    256, 0, stream>>>(pn, out + ROWS * CC + 1);
    k_sim_wmma<<<1,   512, 0, stream>>>(ff, pn, rn, sim);
    k_final  <<<BB,   256, 0, stream>>>(ff, sim, out, am);
    k_entropy<<<1,    128, 0, stream>>>(am, out + ROWS * CC);
}